// HyperLayer_87746181857565
// MI455X (gfx1250) — compile-verified
//
#include <hip/hip_runtime.h>

#define TPB 256
#define DIM 1024

// ---------------------------------------------------------------------------
// Zero the output (harness poisons d_out with 0xAA before timing).
// Vectorized b128 stores; scalar tail handled by a separate kernel only when
// out_size % 4 != 0 (never for the 1024x1024 reference output).
// ---------------------------------------------------------------------------
__global__ void hyper_zero4_kernel(float4* __restrict__ p4, int n4) {
    int i = blockIdx.x * blockDim.x + threadIdx.x;
    if (i < n4) p4[i] = make_float4(0.f, 0.f, 0.f, 0.f);
}

__global__ void hyper_zero1_kernel(float* __restrict__ p, int base, int n) {
    int i = base + (int)threadIdx.x;
    if (i < n) p[i] = 0.f;
}

// ---------------------------------------------------------------------------
// Main scatter kernel: one point per thread per tile; tiles of 256 points are
// staged into LDS with CDNA5 async global->LDS loads (non-temporal: the
// index/value stream is single-use, keep L2 for x/y), double-buffered so the
// next tile's stream is in flight while the current tile computes.
//
// Per point (factorized multilinear form):
//   weight(16 corners) = wA(dims 0,1) * wB(dims 2,3); rows depend only on
//   dims 0-1, cols only on dims 2-3, so:
//     s = sum over 4 input corners  wA_r * x[row_r]     (4 L2 gathers)
//     y[col_c] += v * s * wB_c for 4 output corners     (4 f32 L2 atomics)
//   == the reference's 16-corner gather/scatter, at 1/4 the memory ops.
// ---------------------------------------------------------------------------
__global__ __launch_bounds__(TPB) void hyper_scatter_kernel(
    const float* __restrict__ x,      // [DIM*DIM]
    const float* __restrict__ ridx,   // [N,4]
    const float* __restrict__ rval,   // [N]
    float* __restrict__ y,            // [DIM*DIM]
    int N, int numTiles) {

    __shared__ float4 sIdx[2][TPB];   // 8 KB
    __shared__ float  sVal[2][TPB];   // 2 KB

    const int tid = threadIdx.x;

    // LDS byte offsets for this lane's slots (low 32 bits of generic pointer
    // to addrspace(3) == LDS offset; hardware adds LDS_BASE).
    const uint32_t ldsIdx[2] = { (uint32_t)(uint64_t)&sIdx[0][tid],
                                 (uint32_t)(uint64_t)&sIdx[1][tid] };
    const uint32_t ldsVal[2] = { (uint32_t)(uint64_t)&sVal[0][tid],
                                 (uint32_t)(uint64_t)&sVal[1][tid] };

    auto issue_tile = [&](int tile, int buf) {
        int i = tile * TPB + tid;
        if (i > N - 1) i = N - 1;               // clamp OOB lanes to a valid addr
        uint32_t goffIdx = (uint32_t)i * 16u;   // byte offset into ridx
        uint32_t goffVal = (uint32_t)i * 4u;    // byte offset into rval
        // GVS mode: mem_addr = SGPR64 + VGPR32 offset. Tracked on ASYNCcnt.
        // NT hint: single-use stream, don't displace x/y from L2.
        asm volatile("global_load_async_to_lds_b128 %0, %1, %2 th:TH_LOAD_NT"
                     :: "v"(ldsIdx[buf]), "v"(goffIdx), "s"(ridx) : "memory");
        asm volatile("global_load_async_to_lds_b32 %0, %1, %2 th:TH_LOAD_NT"
                     :: "v"(ldsVal[buf]), "v"(goffVal), "s"(rval) : "memory");
    };

    int tile = blockIdx.x;
    if (tile >= numTiles) return;

    issue_tile(tile, 0);                        // prologue: stage first tile
    int buf = 0;

    for (; tile < numTiles; tile += gridDim.x) {
        int nextTile = tile + (int)gridDim.x;
        if (nextTile < numTiles) {
            issue_tile(nextTile, buf ^ 1);      // prefetch next tile (2 async ops)
            // Async loads retire in order: <=2 outstanding means the current
            // tile's 2 ops are done while the prefetch stays in flight.
            asm volatile("s_wait_asynccnt 2" ::: "memory");
        } else {
            asm volatile("s_wait_asynccnt 0" ::: "memory");
        }

        // Each lane reads only the LDS slots its own wave's async ops filled
        // (ASYNCcnt is per-wave), so no workgroup barrier is required.
        float4 r = sIdx[buf][tid];
        float  v = sVal[buf][tid];

        int i = tile * TPB + tid;
        if (i < N) {
            // Per-dim floor/ceil corner weights: w = 1 - |corner - r|.
            float f0 = floorf(r.x), c0 = ceilf(r.x);
            float f1 = floorf(r.y), c1 = ceilf(r.y);
            float f2 = floorf(r.z), c2 = ceilf(r.z);
            float f3 = floorf(r.w), c3 = ceilf(r.w);
            float wf0 = 1.f - (r.x - f0), wc0 = 1.f - (c0 - r.x);
            float wf1 = 1.f - (r.y - f1), wc1 = 1.f - (c1 - r.y);
            float wf2 = 1.f - (r.z - f2), wc2 = 1.f - (c2 - r.z);
            float wf3 = 1.f - (r.w - f3), wc3 = 1.f - (c3 - r.w);

            // Input gather (dims 0,1): 4 corners, x resident in 192MB L2.
            int a0 = (int)f0 * DIM, b0 = (int)c0 * DIM;
            int a1 = (int)f1,       b1 = (int)c1;
            float s = wf0 * (wf1 * x[a0 + a1] + wc1 * x[a0 + b1])
                    + wc0 * (wf1 * x[b0 + a1] + wc1 * x[b0 + b1]);

            float vs = v * s;

            // Output scatter (dims 2,3): 4 f32 atomics into L2-resident y.
            int a2 = (int)f2 * DIM, b2 = (int)c2 * DIM;
            int a3 = (int)f3,       b3 = (int)c3;
            atomicAdd(&y[a2 + a3], vs * wf2 * wf3);
            atomicAdd(&y[a2 + b3], vs * wf2 * wc3);
            atomicAdd(&y[b2 + a3], vs * wc2 * wf3);
            atomicAdd(&y[b2 + b3], vs * wc2 * wc3);
        }
        buf ^= 1;
    }
}

// ---------------------------------------------------------------------------
// Launch
// ---------------------------------------------------------------------------
extern "C" void kernel_launch(void* const* d_in, const int* in_sizes, int n_in,
                              void* d_out, int out_size, void* d_ws, size_t ws_size,
                              hipStream_t stream) {
    const float* x    = (const float*)d_in[0];   // [1024*1024]
    const float* ridx = (const float*)d_in[1];   // [N,4]
    const float* rval = (const float*)d_in[2];   // [N]
    float*       y    = (float*)d_out;           // [1024*1024]
    const int N = in_sizes[2];

    // 1) zero output (vectorized; out_size for the reference is 4-aligned)
    int n4 = out_size / 4;
    if (n4 > 0) {
        int zblocks = (n4 + TPB - 1) / TPB;
        hyper_zero4_kernel<<<zblocks, TPB, 0, stream>>>((float4*)y, n4);
    }
    if (out_size % 4 != 0) {
        hyper_zero1_kernel<<<1, TPB, 0, stream>>>(y, n4 * 4, out_size);
    }

    // 2) scatter: cap grid so each block loops over >=2 tiles and the async
    //    double-buffering actually overlaps stream-in with compute.
    int tiles = (N + TPB - 1) / TPB;
    int grid  = tiles < 480 ? tiles : 480;
    hyper_scatter_kernel<<<grid, TPB, 0, stream>>>(x, ridx, rval, y, N, tiles);
}